// MemoryModule_3994319585530
// MI455X (gfx1250) — compile-verified
//
#include <hip/hip_runtime.h>

// fp32 WMMA 16x16x4: A(16x4)=v2f, B(4x16)=v2f, C/D(16x16)=v8f, wave32.
typedef __attribute__((ext_vector_type(2))) float v2f;
typedef __attribute__((ext_vector_type(8))) float v8f;

#define WMMA_F32(a, b, c) \
  __builtin_amdgcn_wmma_f32_16x16x4_f32(false, (a), false, (b), (short)0, (c), false, false)

constexpr int Bb = 4, Tt = 4, CK = 64, CV = 128, HW = 48 * 48;  // HW = 2304
constexpr int OC = Tt * CK + CV;                                 // 384 output channels

// ---------------------------------------------------------------------------
// Async global->LDS (128-bit) staging: direct DMA, tracked by ASYNCcnt.
// Builtin signature (from clang diagnostic): (int4 AS1*, int4 AS3*, Imm, Imm).
// ---------------------------------------------------------------------------
typedef int v4i __attribute__((vector_size(16)));
typedef __attribute__((address_space(1))) v4i* gv4i_p;   // global int4*
typedef __attribute__((address_space(3))) v4i* lv4i_p;   // LDS int4*
typedef __attribute__((address_space(3))) void* as3_ptr;

__device__ __forceinline__ void async_b128(const float* gp, float* lp) {
#if __has_builtin(__builtin_amdgcn_global_load_async_to_lds_b128)
  __builtin_amdgcn_global_load_async_to_lds_b128((gv4i_p)gp, (lv4i_p)lp, 0, 0);
#else
  unsigned l32 = (unsigned)(unsigned long long)(as3_ptr)lp;
  asm volatile("global_load_async_to_lds_b128 %0, %1, off"
               :: "v"(l32), "v"(gp) : "memory");
#endif
}

__device__ __forceinline__ void wait_async0() {
#if __has_builtin(__builtin_amdgcn_s_wait_asynccnt)
  __builtin_amdgcn_s_wait_asynccnt(0);
#else
  asm volatile("s_wait_asynccnt 0x0" ::: "memory");
#endif
}

__device__ __forceinline__ void wait_ds0() {
#if __has_builtin(__builtin_amdgcn_s_wait_dscnt)
  __builtin_amdgcn_s_wait_dscnt(0);
#else
  asm volatile("s_wait_dscnt 0x0" ::: "memory");
#endif
}

// ---------------------------------------------------------------------------
// Kernel A: fused flash-attention memory read.
//   S^T = K_chunk(16k x 64c) * Q(64c x 16q)    (D: rows=key in VGPR, cols=query in lane)
//   online softmax over keys (per query column, lane-half reduction via shfl_xor 16)
//   O^T += V^T_chunk(Cv x 16k) * P(16k x 16q)  (D: rows=Cv, cols=query)
// One wave per 16-query tile. V chunk staged via async DMA that overlaps the
// 16 S-tile WMMAs (which read only registers/global).
// ---------------------------------------------------------------------------
#define VP 20  // V-tile row pad (floats): 20*4B = 80B, 16B-aligned, 20*l mod 64 injective

__global__ __launch_bounds__(32) void attn_kernel(const float* __restrict__ mk,
                                                  const float* __restrict__ mv,
                                                  const float* __restrict__ qk,
                                                  float* __restrict__ out) {
  __shared__ float pS[16 * 16];    // P relayout buffer (D-layout -> B-layout)
  __shared__ float vS[128 * VP];   // V chunk, padded rows (bank-conflict free, 16B aligned)

  const int lane = threadIdx.x;
  const int q    = lane & 15;      // query column owned by this lane
  const int half = lane >> 4;      // K-dim half select (A/B operand layout)
  const int m0   = blockIdx.x * 16;
  const int b    = blockIdx.y;

  // Q tile held in registers for the whole key loop: 16 B-operands (4 Ck rows each).
  v2f bq[16];
  {
    const float* qkb = qk + (size_t)b * CK * HW + m0 + q;
    #pragma unroll
    for (int i = 0; i < 16; ++i) {
      int c0 = 4 * i + 2 * half;
      bq[i].x = qkb[(size_t)c0 * HW];
      bq[i].y = qkb[(size_t)(c0 + 1) * HW];
    }
  }

  v8f o[8];                        // O^T accumulators: 8 Cv tiles x (16x16)
  #pragma unroll
  for (int vt = 0; vt < 8; ++vt) o[vt] = (v8f){0, 0, 0, 0, 0, 0, 0, 0};
  float run_max = -__builtin_inff();
  float run_sum = 0.0f;

  for (int t = 0; t < Tt; ++t) {
    const float* mkt = mk + (size_t)(b * Tt + t) * CK * HW;
    const float* mvt = mv + (size_t)(b * Tt + t) * CV * HW;
    for (int n0 = 0; n0 < HW; n0 += 16) {
      wait_ds0();       // previous chunk's LDS reads must land before DMA overwrites
      __syncthreads();
      // Async-stage V chunk (128 x 16 f32) straight into LDS: 16 B128 DMAs/lane.
      #pragma unroll
      for (int s = 0; s < 16; ++s) {
        int quad = lane + 32 * s;      // 0..511 (4-float groups)
        int v    = quad >> 2;
        int kk   = (quad & 3) * 4;
        async_b128(mvt + (size_t)v * HW + n0 + kk, &vS[v * VP + kk]);
      }
      // S^T tile: 16 chained fp32 WMMAs over Ck=64 (overlaps the V DMA).
      v8f s8 = (v8f){0, 0, 0, 0, 0, 0, 0, 0};
      const float* mka = mkt + n0 + q;  // A-operand: lane = key row (n0 + q)
      __builtin_prefetch(mka + 16, 0, 1);  // head start on next key chunk
      #pragma unroll
      for (int i = 0; i < 16; ++i) {
        int c0 = 4 * i + 2 * half;
        v2f a;
        a.x = mka[(size_t)c0 * HW];
        a.y = mka[(size_t)(c0 + 1) * HW];
        s8 = WMMA_F32(a, bq[i], s8);
      }
      // Online softmax for this 16-key chunk. Lane l holds keys {r+8*half}, query q.
      float cmax = s8[0];
      #pragma unroll
      for (int r = 1; r < 8; ++r) cmax = fmaxf(cmax, s8[r]);
      cmax = fmaxf(cmax, __shfl_xor(cmax, 16, 32));          // combine lane halves
      float nmax  = fmaxf(run_max, cmax);
      float alpha = __expf(run_max - nmax);
      float csum  = 0.0f;
      #pragma unroll
      for (int r = 0; r < 8; ++r) {
        float p = __expf(s8[r] - nmax);
        pS[(r + 8 * half) * 16 + q] = p;                     // D-layout -> LDS [key][query]
        csum += p;
      }
      csum += __shfl_xor(csum, 16, 32);
      run_sum = run_sum * alpha + csum;
      run_max = nmax;
      #pragma unroll
      for (int vt = 0; vt < 8; ++vt)
        #pragma unroll
        for (int r = 0; r < 8; ++r) o[vt][r] *= alpha;       // rescale running output
      wait_async0();    // V DMA complete before consuming vS
      __syncthreads();
      // Reload P in B-operand layout (K-dim = key rows).
      v2f bp[4];
      #pragma unroll
      for (int i = 0; i < 4; ++i) {
        bp[i].x = pS[(4 * i + 2 * half) * 16 + q];
        bp[i].y = pS[(4 * i + 1 + 2 * half) * 16 + q];
      }
      // O^T += V^T * P : 8 Cv tiles x 4 K-steps.
      #pragma unroll
      for (int vt = 0; vt < 8; ++vt) {
        #pragma unroll
        for (int i = 0; i < 4; ++i) {
          v2f a;  // A-operand: lane = Cv row, elements = key cols {4i+2h, 4i+1+2h}
          a.x = vS[(vt * 16 + q) * VP + 4 * i + 2 * half];
          a.y = vS[(vt * 16 + q) * VP + 4 * i + 1 + 2 * half];
          o[vt] = WMMA_F32(a, bp[i], o[vt]);
        }
      }
    }
  }

  // Normalize and write memory output to channels [256, 384).
  float inv = 1.0f / run_sum;
  float* ob = out + ((size_t)b * OC + Tt * CK) * HW + m0 + q;
  #pragma unroll
  for (int vt = 0; vt < 8; ++vt)
    #pragma unroll
    for (int r = 0; r < 8; ++r)
      ob[(size_t)(vt * 16 + r + 8 * half) * HW] = o[vt][r] * inv;
}

// ---------------------------------------------------------------------------
// Kernel B1: per-(b,t) gram  G = mk(64 x HW) * mv(128 x HW)^T  via fp32 WMMA.
// One WG per (b,t); wave w owns value-tile w, accumulating 4 Ck tiles.
// Slabs staged with async B128 DMA into 36-padded LDS rows.
// ---------------------------------------------------------------------------
#define GP 36  // 36*4B = 144B rows (16B-aligned); 36*l mod 64 injective over 16 lanes

__global__ __launch_bounds__(256) void gram_kernel(const float* __restrict__ mk,
                                                   const float* __restrict__ mv,
                                                   float* __restrict__ g) {
  __shared__ float mkS[64 * GP];
  __shared__ float mvS[128 * GP];
  const int bt   = blockIdx.x;
  const int tid  = threadIdx.x;
  const int lane = tid & 31;
  const int wave = tid >> 5;        // value tile index 0..7
  const int q    = lane & 15;
  const int half = lane >> 4;

  const float* mkb = mk + (size_t)bt * CK * HW;
  const float* mvb = mv + (size_t)bt * CV * HW;

  v8f acc[4];
  #pragma unroll
  for (int ct = 0; ct < 4; ++ct) acc[ct] = (v8f){0, 0, 0, 0, 0, 0, 0, 0};

  for (int n0 = 0; n0 < HW; n0 += 32) {
    wait_ds0();
    __syncthreads();
    // mk slab: 64 x 32 f32 = 512 quads; mv slab: 128 x 32 = 1024 quads.
    #pragma unroll
    for (int s = 0; s < 2; ++s) {
      int idx = tid + 256 * s;
      int c   = idx >> 3;
      int kk  = (idx & 7) * 4;
      async_b128(mkb + (size_t)c * HW + n0 + kk, &mkS[c * GP + kk]);
    }
    #pragma unroll
    for (int s = 0; s < 4; ++s) {
      int idx = tid + 256 * s;
      int v   = idx >> 3;
      int kk  = (idx & 7) * 4;
      async_b128(mvb + (size_t)v * HW + n0 + kk, &mvS[v * GP + kk]);
    }
    wait_async0();
    __syncthreads();
    #pragma unroll
    for (int i = 0; i < 8; ++i) {
      v2f bv;  // B-operand: K = n rows, lane = Cv column
      bv.x = mvS[(wave * 16 + q) * GP + 4 * i + 2 * half];
      bv.y = mvS[(wave * 16 + q) * GP + 4 * i + 1 + 2 * half];
      #pragma unroll
      for (int ct = 0; ct < 4; ++ct) {
        v2f a;  // A-operand: lane = Ck row, elements = n cols
        a.x = mkS[(ct * 16 + q) * GP + 4 * i + 2 * half];
        a.y = mkS[(ct * 16 + q) * GP + 4 * i + 1 + 2 * half];
        acc[ct] = WMMA_F32(a, bv, acc[ct]);
      }
    }
  }
  float* gb = g + (size_t)bt * CK * CV;
  #pragma unroll
  for (int ct = 0; ct < 4; ++ct)
    #pragma unroll
    for (int r = 0; r < 8; ++r)
      gb[(size_t)(ct * 16 + r + 8 * half) * CV + wave * 16 + q] = acc[ct][r];
}

// ---------------------------------------------------------------------------
// Kernel B2: softmax over the T axis of the gram (4 values per (b,c,v)).
// ---------------------------------------------------------------------------
__global__ void softmax_t_kernel(float* __restrict__ g) {
  int i  = blockIdx.x * blockDim.x + threadIdx.x;  // B*CK*CV = 32768
  int b  = i >> 13;
  int cv = i & 8191;
  float x[Tt];
  #pragma unroll
  for (int t = 0; t < Tt; ++t) x[t] = g[(size_t)(b * Tt + t) * (CK * CV) + cv];
  float mx = x[0];
  #pragma unroll
  for (int t = 1; t < Tt; ++t) mx = fmaxf(mx, x[t]);
  float sum = 0.0f;
  #pragma unroll
  for (int t = 0; t < Tt; ++t) { x[t] = __expf(x[t] - mx); sum += x[t]; }
  float inv = 1.0f / sum;
  #pragma unroll
  for (int t = 0; t < Tt; ++t) g[(size_t)(b * Tt + t) * (CK * CV) + cv] = x[t] * inv;
}

// ---------------------------------------------------------------------------
// Kernel B3: qv_out[b,t] = G[b,t](64x128) @ qv[b](128 x HW) via fp32 WMMA.
// Whole G async-staged into 132-padded LDS; qv B-operands (K=128) kept in
// registers and reused across the 4 Ck tiles.
// ---------------------------------------------------------------------------
#define QP 132  // 132*4B = 528B rows (16B-aligned); 132*l mod 64 = 4l, injective

__global__ __launch_bounds__(256) void qvout_kernel(const float* __restrict__ g,
                                                    const float* __restrict__ qv,
                                                    float* __restrict__ out) {
  __shared__ float gS[64 * QP];
  const int bt   = blockIdx.y;
  const int b    = bt >> 2;
  const int t    = bt & 3;
  const int tid  = threadIdx.x;
  const int lane = tid & 31;
  const int wave = tid >> 5;
  const int q    = lane & 15;
  const int half = lane >> 4;
  const int m0   = (blockIdx.x * 8 + wave) * 16;

  // Stage G (64 x 128 f32 = 2048 quads) via async DMA.
  const float* gbase = g + (size_t)bt * (CK * CV);
  #pragma unroll
  for (int s = 0; s < 8; ++s) {
    int idx = tid + 256 * s;       // 0..2047
    int c   = idx >> 5;
    int v4  = (idx & 31) * 4;
    async_b128(gbase + (size_t)c * CV + v4, &gS[c * QP + v4]);
  }

  // qv tile as 32 B-operands covering K = Cv = 128 (overlaps the G DMA).
  v2f bqv[32];
  const float* qvb = qv + (size_t)b * CV * HW + m0 + q;
  __builtin_prefetch(qvb, 0, 1);
  #pragma unroll
  for (int i = 0; i < 32; ++i) {
    int v0 = 4 * i + 2 * half;
    bqv[i].x = qvb[(size_t)v0 * HW];
    bqv[i].y = qvb[(size_t)(v0 + 1) * HW];
  }

  wait_async0();
  __syncthreads();

  float* ob = out + ((size_t)b * OC + t * CK) * HW + m0 + q;
  for (int ct = 0; ct < 4; ++ct) {
    v8f acc = (v8f){0, 0, 0, 0, 0, 0, 0, 0};
    #pragma unroll
    for (int i = 0; i < 32; ++i) {
      v2f a;  // A-operand: lane = Ck row of G, elements = Cv cols
      a.x = gS[(ct * 16 + q) * QP + 4 * i + 2 * half];
      a.y = gS[(ct * 16 + q) * QP + 4 * i + 1 + 2 * half];
      acc = WMMA_F32(a, bqv[i], acc);
    }
    #pragma unroll
    for (int r = 0; r < 8; ++r)
      ob[(size_t)(ct * 16 + r + 8 * half) * HW] = acc[r];
  }
}

// ---------------------------------------------------------------------------
extern "C" void kernel_launch(void* const* d_in, const int* in_sizes, int n_in,
                              void* d_out, int out_size, void* d_ws, size_t ws_size,
                              hipStream_t stream) {
  const float* mk = (const float*)d_in[0];  // (B,T,CK,H,W)
  const float* mv = (const float*)d_in[1];  // (B,T,CV,H,W)
  const float* qk = (const float*)d_in[2];  // (B,CK,H,W)
  const float* qv = (const float*)d_in[3];  // (B,CV,H,W)
  float* out = (float*)d_out;               // (B, T*CK + CV, H, W)
  float* g   = (float*)d_ws;                // B*T*CK*CV floats = 512 KB scratch

  // Attention read -> out channels [256, 384)
  attn_kernel<<<dim3(HW / 16, Bb), 32, 0, stream>>>(mk, mv, qk, out);
  // Gram -> softmax over T -> remix -> out channels [0, 256)
  gram_kernel<<<dim3(Bb * Tt), 256, 0, stream>>>(mk, mv, g);
  softmax_t_kernel<<<dim3((Bb * CK * CV) / 256), 256, 0, stream>>>(g);
  qvout_kernel<<<dim3(HW / 16 / 8, Bb * Tt), 256, 0, stream>>>(g, qv, out);
}